// GINLayer_1486058684700
// MI455X (gfx1250) — compile-verified
//
#include <hip/hip_runtime.h>

// ---------------------------------------------------------------------------
// GIN layer for MI455X (gfx1250):
//   aggr = scatter_add(x[src] -> dst);  out = (1+eps)*x + aggr
//   h1 = relu(BN(out @ W1 + b1));  h2 = relu(BN(h1 @ W2 + b2))
// GEMMs run on V_WMMA_F32_16X16X4_F32 (fp32 matrix pipe, matches fp32 ref).
// ---------------------------------------------------------------------------

typedef __attribute__((ext_vector_type(2))) float v2f;
typedef __attribute__((ext_vector_type(8))) float v8f;

#define DFEAT 128
#define ROW_TILES 5
#define ROWS_PER_BLOCK (ROW_TILES * 16)   // 80 rows/block; 50000 % 80 == 0
#define LDS_STRIDE (DFEAT + 4)            // pad 4 dwords -> conflict-free DS reads

// ---------------------------- utility kernels ------------------------------

__global__ __launch_bounds__(256) void zero_kernel(float* __restrict__ p, int n) {
    int i = blockIdx.x * 256 + threadIdx.x;
    if (i < n) p[i] = 0.0f;
}

// one thread per (edge, feature): gather x[src], global_atomic_add_f32 to aggr[dst]
__global__ __launch_bounds__(256) void scatter_add_kernel(
    const float* __restrict__ x, const int* __restrict__ ei,
    float* __restrict__ aggr, int E) {
    unsigned tid   = blockIdx.x * 256u + threadIdx.x;
    unsigned total = (unsigned)E * (unsigned)DFEAT;
    if (tid < total) {
        unsigned e = tid >> 7;          // / 128
        unsigned d = tid & (DFEAT - 1);
        int src = ei[e];
        int dst = ei[E + e];
        unsafeAtomicAdd(&aggr[(unsigned)dst * DFEAT + d],
                        x[(unsigned)src * DFEAT + d]);
    }
}

// ------------------------------ WMMA GEMM ----------------------------------
// MODE 0: A-row = (1+eps)*x + aggr          (epsp = &eps scalar)
// MODE 1: A-row = relu(h_prev*scale + shift) (epsp = scale[128], shift[128])
// Computes Hout = A @ W + bias   (A: N x 128, W: 128 x 128 row-major)
template <int MODE>
__global__ __launch_bounds__(256) void gemm_bn_kernel(
    const float* __restrict__ A0, const float* __restrict__ A1,
    const float* __restrict__ epsp, const float* __restrict__ shift,
    const float* __restrict__ W, const float* __restrict__ bias,
    float* __restrict__ Hout, int N) {

    __shared__ float As[ROWS_PER_BLOCK][LDS_STRIDE];

    const int rowBase = blockIdx.x * ROWS_PER_BLOCK;
    const int tid     = threadIdx.x;

    float fe = 0.0f;
    if (MODE == 0) fe = 1.0f + epsp[0];

    // Stage the 80x128 activation tile into LDS, applying the input transform.
    for (int i = tid; i < ROWS_PER_BLOCK * DFEAT; i += 256) {
        int r   = i >> 7;            // / 128
        int c   = i & (DFEAT - 1);
        int row = rowBase + r;
        float v = 0.0f;
        if (row < N) {
            int idx = row * DFEAT + c;
            if (MODE == 0) {
                v = fe * A0[idx] + A1[idx];
            } else {
                v = A0[idx] * epsp[c] + shift[c];
                v = v > 0.0f ? v : 0.0f;
            }
        }
        As[r][c] = v;
    }
    __syncthreads();

    const int wave    = tid >> 5;     // 8 waves -> 8 column tiles of 16
    const int lane    = tid & 31;
    const int colBase = wave * 16;
    const int lrow    = lane & 15;
    const bool hi     = lane >= 16;

    v8f acc[ROW_TILES] = {};          // fp32 16x16 accumulators

    // K loop: 32 steps of V_WMMA_F32_16X16X4_F32 per row tile.
    // A 16x4 fragment: lanes 0-15 hold {K=k0,k0+1}, lanes 16-31 hold {k0+2,k0+3}
    // B 4x16 fragment: analogous split of K across lane halves, N striped on lanes.
    #pragma unroll
    for (int k0 = 0; k0 < DFEAT; k0 += 4) {
        const int kB = k0 + (hi ? 2 : 0);
        v2f bf;
        bf.x = W[(kB + 0) * DFEAT + colBase + lrow];
        bf.y = W[(kB + 1) * DFEAT + colBase + lrow];
        #pragma unroll
        for (int t = 0; t < ROW_TILES; ++t) {
            const int ar = t * 16 + lrow;
            v2f af = *(const v2f*)&As[ar][kB];   // 8B-aligned DS read
            acc[t] = __builtin_amdgcn_wmma_f32_16x16x4_f32(
                false, af, false, bf, (short)0, acc[t], false, false);
        }
    }

    // Store D-matrix: VGPR r <-> row r (lanes 0-15) / row r+8 (lanes 16-31)
    const int col  = colBase + lrow;
    const float bc = bias[col];
    #pragma unroll
    for (int t = 0; t < ROW_TILES; ++t) {
        const int tileRow = rowBase + t * 16 + (hi ? 8 : 0);
        #pragma unroll
        for (int r = 0; r < 8; ++r) {
            int row = tileRow + r;
            if (row < N) Hout[row * DFEAT + col] = acc[t][r] + bc;
        }
    }
}

// ----------------------- batch-norm statistics -----------------------------

__global__ __launch_bounds__(256) void colstats_kernel(
    const float* __restrict__ H, float* __restrict__ sum,
    float* __restrict__ sumsq, int N) {
    __shared__ float ssum[256];
    __shared__ float ssq[256];
    const int c    = threadIdx.x & (DFEAT - 1);
    const int rOff = threadIdx.x >> 7;          // 0 or 1
    float s = 0.0f, q = 0.0f;
    for (int r = blockIdx.x * 2 + rOff; r < N; r += gridDim.x * 2) {
        float v = H[r * DFEAT + c];
        s += v;
        q += v * v;
    }
    ssum[threadIdx.x] = s;
    ssq[threadIdx.x]  = q;
    __syncthreads();
    if (threadIdx.x < 128) {
        s = ssum[threadIdx.x] + ssum[threadIdx.x + 128];
        q = ssq[threadIdx.x] + ssq[threadIdx.x + 128];
        unsafeAtomicAdd(&sum[c], s);
        unsafeAtomicAdd(&sumsq[c], q);
    }
}

__global__ void bn_finalize_kernel(const float* __restrict__ sum,
                                   const float* __restrict__ sumsq,
                                   const float* __restrict__ g,
                                   const float* __restrict__ beta,
                                   float* __restrict__ scale,
                                   float* __restrict__ shift, float invN) {
    int c = threadIdx.x;            // 128 threads
    float mean = sum[c] * invN;
    float var  = sumsq[c] * invN - mean * mean;   // biased variance
    float sc   = g[c] * rsqrtf(var + 1e-5f);
    scale[c]   = sc;
    shift[c]   = beta[c] - mean * sc;
}

__global__ __launch_bounds__(256) void apply_bn_relu_kernel(
    const float* __restrict__ H, const float* __restrict__ scale,
    const float* __restrict__ shift, float* __restrict__ out, int total) {
    int i = blockIdx.x * 256 + threadIdx.x;
    if (i < total) {
        int c   = i & (DFEAT - 1);
        float v = H[i] * scale[c] + shift[c];
        out[i]  = v > 0.0f ? v : 0.0f;
    }
}

// ------------------------------- launcher ----------------------------------

extern "C" void kernel_launch(void* const* d_in, const int* in_sizes, int n_in,
                              void* d_out, int out_size, void* d_ws, size_t ws_size,
                              hipStream_t stream) {
    const float* x     = (const float*)d_in[0];
    const int*   ei    = (const int*)d_in[1];     // edge_index, shape (2, E) flat
    const float* eps   = (const float*)d_in[2];
    const float* W1    = (const float*)d_in[3];
    const float* b1    = (const float*)d_in[4];
    const float* g1    = (const float*)d_in[5];
    const float* beta1 = (const float*)d_in[6];
    const float* W2    = (const float*)d_in[7];
    const float* b2    = (const float*)d_in[8];
    const float* g2    = (const float*)d_in[9];
    const float* beta2 = (const float*)d_in[10];

    const int N = in_sizes[0] / DFEAT;   // 50000
    const int E = in_sizes[1] / 2;       // 800000

    float* ws    = (float*)d_ws;
    float* aggr  = ws;                          // N*128
    float* h1    = ws + (size_t)N * DFEAT;      // N*128
    float* h2    = aggr;                        // reuse: aggr dead after GEMM1
    float* stats = h1 + (size_t)N * DFEAT;      // 8 * 128 floats
    float *sum1 = stats,       *sq1 = stats + 128,
          *scale1 = stats + 256, *shift1 = stats + 384,
          *sum2 = stats + 512,  *sq2 = stats + 640,
          *scale2 = stats + 768, *shift2 = stats + 896;

    const int nAgg = N * DFEAT;
    zero_kernel<<<(nAgg + 255) / 256, 256, 0, stream>>>(aggr, nAgg);
    zero_kernel<<<4, 256, 0, stream>>>(stats, 1024);

    unsigned scatterThreads = (unsigned)E * (unsigned)DFEAT;
    scatter_add_kernel<<<(scatterThreads + 255u) / 256u, 256, 0, stream>>>(
        x, ei, aggr, E);

    const int gblocks = (N + ROWS_PER_BLOCK - 1) / ROWS_PER_BLOCK;

    gemm_bn_kernel<0><<<gblocks, 256, 0, stream>>>(x, aggr, eps, nullptr,
                                                   W1, b1, h1, N);
    colstats_kernel<<<512, 256, 0, stream>>>(h1, sum1, sq1, N);
    bn_finalize_kernel<<<1, 128, 0, stream>>>(sum1, sq1, g1, beta1,
                                              scale1, shift1, 1.0f / (float)N);

    gemm_bn_kernel<1><<<gblocks, 256, 0, stream>>>(h1, nullptr, scale1, shift1,
                                                   W2, b2, h2, N);
    colstats_kernel<<<512, 256, 0, stream>>>(h2, sum2, sq2, N);
    bn_finalize_kernel<<<1, 128, 0, stream>>>(sum2, sq2, g2, beta2,
                                              scale2, shift2, 1.0f / (float)N);

    apply_bn_relu_kernel<<<(N * DFEAT + 255) / 256, 256, 0, stream>>>(
        h2, scale2, shift2, (float*)d_out, N * DFEAT);
}